// TextRecoveryWithVisualContext_34428457845135
// MI455X (gfx1250) — compile-verified
//
#include <hip/hip_runtime.h>
#include <math.h>

typedef __attribute__((ext_vector_type(2))) float v2f;
typedef __attribute__((ext_vector_type(4))) float v4f;
typedef __attribute__((ext_vector_type(8))) float v8f;

// ---- workspace layout (float offsets) ----
#define WS_WCOL   0      // [256]  column sums of visual_proj_w
#define WS_BSUM   256    // [1]    sum of visual_proj_b
#define WS_TEMAX  257    // [1]
#define WS_TEMIN  258    // [1]
#define WS_M      260    // [16]   per-batch softmax max
#define WS_RZ     276    // [16]   per-batch 1/Z
#define WS_S      512    // [16*256] s[b,p]
#define WS_VBAR   4608   // [16*256] mean_hw VF
#define WS_X2     8704   // [16*512] Wv * vbar + bv

// ---------------------------------------------------------------------------
// K1: wcol[c] = sum_d Wv[d,c]; bias_sum; te max/min
// ---------------------------------------------------------------------------
__global__ __launch_bounds__(256) void k_stats0(const float* __restrict__ vw,
                                                const float* __restrict__ vb,
                                                const float* __restrict__ te,
                                                float* __restrict__ ws) {
  __shared__ float red[256];
  const int t = threadIdx.x;

  float acc = 0.f;
  #pragma unroll 8
  for (int d = 0; d < 512; ++d) acc += vw[d * 256 + t];
  ws[WS_WCOL + t] = acc;

  red[t] = vb[t] + vb[t + 256];
  __syncthreads();
  for (int off = 128; off > 0; off >>= 1) {
    if (t < off) red[t] += red[t + off];
    __syncthreads();
  }
  if (t == 0) ws[WS_BSUM] = red[0];
  __syncthreads();

  float mx = -INFINITY, mn = INFINITY;
  for (int j = t; j < 8192; j += 256) {
    float v = te[j];
    mx = fmaxf(mx, v);
    mn = fminf(mn, v);
  }
  red[t] = mx;
  __syncthreads();
  for (int off = 128; off > 0; off >>= 1) {
    if (t < off) red[t] = fmaxf(red[t], red[t + off]);
    __syncthreads();
  }
  if (t == 0) ws[WS_TEMAX] = red[0];
  __syncthreads();
  red[t] = mn;
  __syncthreads();
  for (int off = 128; off > 0; off >>= 1) {
    if (t < off) red[t] = fminf(red[t], red[t + off]);
    __syncthreads();
  }
  if (t == 0) ws[WS_TEMIN] = red[0];
}

// ---------------------------------------------------------------------------
// K2: per batch b -> s[b,p] = sum_c wcol[c]*VF[b,c,p] + bias_sum
//                    vbar[b,c] = mean_p VF[b,c,p]
// ---------------------------------------------------------------------------
__global__ __launch_bounds__(256) void k_sv(const float* __restrict__ vf,
                                            float* __restrict__ ws) {
  __shared__ float wc[256];
  const int b = blockIdx.x, t = threadIdx.x;
  wc[t] = ws[WS_WCOL + t];
  __syncthreads();

  const float* vfb = vf + (size_t)b * 256 * 256;

  float acc = ws[WS_BSUM];
  #pragma unroll 4
  for (int c = 0; c < 256; ++c) acc += wc[c] * vfb[c * 256 + t];
  ws[WS_S + b * 256 + t] = acc;

  const float* row = vfb + t * 256;
  float a2 = 0.f;
  #pragma unroll 8
  for (int p = 0; p < 256; ++p) a2 += row[p];
  ws[WS_VBAR + b * 256 + t] = a2 * (1.0f / 256.0f);
}

// ---------------------------------------------------------------------------
// K3: per batch b -> softmax max m[b] (separable outer product) and 1/Z[b]
// ---------------------------------------------------------------------------
__global__ __launch_bounds__(256) void k_softstats(const float* __restrict__ te,
                                                   float* __restrict__ ws) {
  __shared__ float tel[8192];   // 32 KB
  __shared__ float red[256];
  __shared__ float mshare;
  const int b = blockIdx.x, t = threadIdx.x;

  for (int j = t; j < 8192; j += 256) tel[j] = te[j];

  const float sp   = ws[WS_S + b * 256 + t];
  const float tmax = ws[WS_TEMAX];
  const float tmin = ws[WS_TEMIN];
  red[t] = (sp >= 0.f) ? sp * tmax : sp * tmin;
  __syncthreads();
  for (int off = 128; off > 0; off >>= 1) {
    if (t < off) red[t] = fmaxf(red[t], red[t + off]);
    __syncthreads();
  }
  if (t == 0) mshare = red[0];
  __syncthreads();
  const float m = mshare;

  float z = 0.f;
  #pragma unroll 4
  for (int j = 0; j < 8192; ++j) z += __expf(tel[j] * sp - m);
  __syncthreads();
  red[t] = z;
  __syncthreads();
  for (int off = 128; off > 0; off >>= 1) {
    if (t < off) red[t] += red[t + off];
    __syncthreads();
  }
  if (t == 0) {
    ws[WS_M + b]  = m;
    ws[WS_RZ + b] = 1.0f / red[0];
  }
}

// ---------------------------------------------------------------------------
// K4: attn[b,n,d,p] = exp(te[n,d]*s[b,p] - m[b]) / Z[b]
// Dominant kernel: 134 MB write-once stream. Each thread owns 4 consecutive
// spatial positions -> one b128 non-temporal store per (d) step. te row
// cached in LDS (broadcast reads).
// ---------------------------------------------------------------------------
__global__ __launch_bounds__(256) void k_attn(const float* __restrict__ te,
                                              const float* __restrict__ ws,
                                              float* __restrict__ attn) {
  __shared__ float tel[512];
  const int n = blockIdx.x, b = blockIdx.y, t = threadIdx.x;
  tel[t]       = te[n * 512 + t];
  tel[t + 256] = te[n * 512 + t + 256];
  __syncthreads();

  const int p0 = (t & 63) * 4;        // 4 consecutive spatial positions
  const int d0 = (t >> 6) * 128;      // 128-wide d slab per thread group
  const v4f sp = *(const v4f*)(ws + WS_S + b * 256 + p0);
  const float m  = ws[WS_M + b];
  const float rz = ws[WS_RZ + b];

  float* base = attn + ((size_t)(b * 16 + n) * 512 + d0) * 256 + p0;
  #pragma unroll 4
  for (int i = 0; i < 128; ++i) {
    const float tv = tel[d0 + i];
    v4f r;
    r.x = __expf(tv * sp.x - m) * rz;
    r.y = __expf(tv * sp.y - m) * rz;
    r.z = __expf(tv * sp.z - m) * rz;
    r.w = __expf(tv * sp.w - m) * rz;
    __builtin_nontemporal_store(r, (v4f*)(base + (size_t)i * 256));
  }
}

// ---------------------------------------------------------------------------
// K5a: x2[b,d] = sum_c Wv[d,c]*vbar[b,c] + bv[d]   (WMMA f32 16x16x4, K=256)
// One wave (32 threads) per 16-column tile of d.
// ---------------------------------------------------------------------------
__global__ __launch_bounds__(32) void k_x2(const float* __restrict__ vw,
                                           const float* __restrict__ vb,
                                           float* __restrict__ ws) {
  const int tile = blockIdx.x;
  const int lane = threadIdx.x;
  const int mrow  = lane & 15;            // A row (batch)
  const int khalf = (lane >> 4) * 2;      // K sub-offset per ISA A/B layout
  const int ncol  = tile * 16 + (lane & 15);  // B column (output d)

  const float* A = ws + WS_VBAR;          // [16 x 256], row-major
  v8f c = {0.f, 0.f, 0.f, 0.f, 0.f, 0.f, 0.f, 0.f};

  #pragma unroll 4
  for (int kb = 0; kb < 256; kb += 4) {
    const int k0 = kb + khalf;            // even -> 8B aligned
    v2f a = *(const v2f*)(A + mrow * 256 + k0);
    v2f bm = *(const v2f*)(vw + (size_t)ncol * 256 + k0);
    c = __builtin_amdgcn_wmma_f32_16x16x4_f32(false, a, false, bm,
                                              (short)0, c, false, false);
  }

  const float bias = vb[ncol];
  #pragma unroll
  for (int v = 0; v < 8; ++v) {
    const int row = v + 8 * (lane >> 4);  // D-matrix row layout
    ws[WS_X2 + row * 512 + ncol] = c[v] + bias;
  }
}

// ---------------------------------------------------------------------------
// K5b: final[b,d] = sum_d' iW[d,d']*mtf[b,d'] + sum_d' iW[d,512+d']*x2[b,d'] + ib[d]
// Two K=512 WMMA sweeps into one accumulator.
// ---------------------------------------------------------------------------
__global__ __launch_bounds__(32) void k_final(const float* __restrict__ mtf,
                                              const float* __restrict__ iw,
                                              const float* __restrict__ ib,
                                              const float* __restrict__ ws,
                                              float* __restrict__ out) {
  const int tile = blockIdx.x;
  const int lane = threadIdx.x;
  const int mrow  = lane & 15;
  const int khalf = (lane >> 4) * 2;
  const int ncol  = tile * 16 + (lane & 15);

  const float* x2 = ws + WS_X2;           // [16 x 512]
  v8f c = {0.f, 0.f, 0.f, 0.f, 0.f, 0.f, 0.f, 0.f};

  #pragma unroll 4
  for (int kb = 0; kb < 512; kb += 4) {   // recovered-text half of interaction_w
    const int k0 = kb + khalf;
    v2f a = *(const v2f*)(mtf + mrow * 512 + k0);
    v2f bm = *(const v2f*)(iw + (size_t)ncol * 1024 + k0);
    c = __builtin_amdgcn_wmma_f32_16x16x4_f32(false, a, false, bm,
                                              (short)0, c, false, false);
  }
  #pragma unroll 4
  for (int kb = 0; kb < 512; kb += 4) {   // visual half of interaction_w
    const int k0 = kb + khalf;
    v2f a = *(const v2f*)(x2 + mrow * 512 + k0);
    v2f bm = *(const v2f*)(iw + (size_t)ncol * 1024 + 512 + k0);
    c = __builtin_amdgcn_wmma_f32_16x16x4_f32(false, a, false, bm,
                                              (short)0, c, false, false);
  }

  const float bias = ib[ncol];
  #pragma unroll
  for (int v = 0; v < 8; ++v) {
    const int row = v + 8 * (lane >> 4);
    out[row * 512 + ncol] = c[v] + bias;
  }
}

// ---------------------------------------------------------------------------
extern "C" void kernel_launch(void* const* d_in, const int* in_sizes, int n_in,
                              void* d_out, int out_size, void* d_ws, size_t ws_size,
                              hipStream_t stream) {
  const float* mtf = (const float*)d_in[0];   // [16,512]
  const float* vf  = (const float*)d_in[1];   // [16,256,16,16]
  const float* te  = (const float*)d_in[2];   // [16,512]
  const float* vw  = (const float*)d_in[3];   // [512,256]
  const float* vb  = (const float*)d_in[4];   // [512]
  const float* iw  = (const float*)d_in[5];   // [512,1024]
  const float* ib  = (const float*)d_in[6];   // [512]

  float* out  = (float*)d_out;                // final: [16,512]
  float* attn = out + 16 * 512;               // attn: [16,16,512,16,16]
  float* ws   = (float*)d_ws;

  k_stats0   <<<1,            256, 0, stream>>>(vw, vb, te, ws);
  k_sv       <<<16,           256, 0, stream>>>(vf, ws);
  k_softstats<<<16,           256, 0, stream>>>(te, ws);
  k_attn     <<<dim3(16, 16), 256, 0, stream>>>(te, ws, attn);
  k_x2       <<<32,           32,  0, stream>>>(vw, vb, ws);
  k_final    <<<32,           32,  0, stream>>>(mtf, iw, ib, ws, out);
}